// OpacityApproximator_6141803233761
// MI455X (gfx1250) — compile-verified
//
#include <hip/hip_runtime.h>
#include <cstdint>
#include <cstddef>

// ---------------------------------------------------------------------------
// VolSDF error-bounded up-sampling for MI455X (gfx1250, wave32).
// One wave per ray; per-ray sample arrays double-buffered in LDS; f32 wave
// scans via shuffles; wave-wide f32 sum via V_WMMA_F32_16X16X4_F32; final
// LDS->HBM tile stores via the Tensor Data Mover (tensor_store_from_lds).
// ---------------------------------------------------------------------------

namespace {
constexpr int   kN            = 64;      // samples added per iteration
constexpr float kEps          = 0.1f;
constexpr int   kAlgoIters    = 5;
constexpr int   kBisect       = 10;
constexpr int   kNp           = 16384;
constexpr int   kMaxNs        = 384;     // 64 + 5*64
constexpr int   kMaxCh        = 12;      // ceil(383/32)
constexpr int   kRaysPerBlock = 8;
constexpr int   kThreads      = kRaysPerBlock * 32;
}

typedef float        v2f  __attribute__((ext_vector_type(2)));
typedef float        v8f  __attribute__((ext_vector_type(8)));
typedef unsigned int v4u  __attribute__((ext_vector_type(4)));
typedef int          v4i  __attribute__((ext_vector_type(4)));
typedef int          v8i  __attribute__((ext_vector_type(8)));

// ---------------- wave32 primitives ----------------------------------------

__device__ __forceinline__ float wave_excl_scan_f(float v, int lane) {
  float x = v;
#pragma unroll
  for (int off = 1; off < 32; off <<= 1) {
    float y = __shfl_up(x, off, 32);
    if (lane >= off) x += y;
  }
  return x - v;
}

__device__ __forceinline__ int wave_excl_scan_i(int v, int lane) {
  int x = v;
#pragma unroll
  for (int off = 1; off < 32; off <<= 1) {
    int y = __shfl_up(x, off, 32);
    if (lane >= off) x += y;
  }
  return x - v;
}

__device__ __forceinline__ int wave_sum_i(int v) {
#pragma unroll
  for (int off = 16; off >= 1; off >>= 1) v += __shfl_xor(v, off, 32);
  return v;
}

__device__ __forceinline__ float wave_max_f(float v) {
#pragma unroll
  for (int off = 16; off >= 1; off >>= 1) v = fmaxf(v, __shfl_xor(v, off, 32));
  return v;
}

// Wave-wide f32 sum through the matrix unit.
// D = A(16x4, all ones) * B(4x16, one lane value per storage slot) gives
// colsum[n] replicated in every row of D. Because A is all-ones, the result
// is independent of the exact B element<->slot bijection: sum of all column
// sums == sum of all 64 B slots == sum of the 32 lane values (rest are 0).
// D VGPR0 holds row0 (lanes 0-15) / row8 (lanes 16-31) = colsum[lane%16];
// a 16-wide butterfly finishes the reduction exactly in f32.
__device__ __forceinline__ float wave_sum_wmma(float t) {
  v2f a; a[0] = 1.0f; a[1] = 1.0f;   // A = ones (covers all 64 A slots)
  v2f b; b[0] = t;    b[1] = 0.0f;   // 32 data slots + 32 zeros
  v8f c = {};
  v8f d = __builtin_amdgcn_wmma_f32_16x16x4_f32(
      /*neg_a=*/false, a, /*neg_b=*/false, b,
      /*c_mod=*/(short)0, c, /*reuse_a=*/false, /*reuse_b=*/false);
  float p = d[0];
  p += __shfl_xor(p, 1, 32);
  p += __shfl_xor(p, 2, 32);
  p += __shfl_xor(p, 4, 32);
  p += __shfl_xor(p, 8, 32);
  return p;  // total broadcast to all lanes
}

// ---------------- VolSDF math ----------------------------------------------

__device__ __forceinline__ float vol_sigma(float s, float b) {
  float a = 1.0f / b;
  float tmp = 0.5f * expf(-fabsf(s * a));
  return a * ((s <= 0.0f) ? tmp : (1.0f - tmp));
}

// Evaluate approx_error_intervals for one ray (wave-cooperative).
// STORE=false: return max over intervals (find_beta_p / bisection path).
// STORE=true : write min(err,100) per owned interval into `out` and return
//              the ray-wide sum (upsample path; sum goes through WMMA).
template <bool STORE>
__device__ __forceinline__ float eval_error(
    float b, int NI, int CH, int base, int lane,
    const float* dlt, const float* s0c, const float* dsc, float* out) {
  float x1[kMaxCh], x2[kMaxCh];
  float t1 = 0.0f, t2 = 0.0f;
#pragma unroll
  for (int j = 0; j < kMaxCh; ++j) {
    int i = base + j;
    bool ok = (j < CH) && (i < NI);
    float d  = ok ? dlt[j] : 0.0f;
    float sg = ok ? vol_sigma(s0c[j], b) : 0.0f;
    float e2 = ok ? expf(-dsc[j] / b) : 0.0f;
    x1[j] = d * sg;        // delta * sigma[:-1]
    x2[j] = d * d * e2;    // delta^2 * exp(-d_star/b)
    t1 += x1[j];
    t2 += x2[j];
  }
  float acc1 = wave_excl_scan_f(t1, lane);  // exclusive prefix of x1 (== R)
  float acc2 = wave_excl_scan_f(t2, lane);  // running prefix of x2
  float inv4b2 = 0.25f / (b * b);
  float m = -3.0e38f;
  float ssum = 0.0f;
#pragma unroll
  for (int j = 0; j < kMaxCh; ++j) {
    int i = base + j;
    if ((j < CH) && (i < NI)) {
      float R = acc1;          // exclusive cumsum of x1 at i
      acc1 += x1[j];
      acc2 += x2[j];           // inclusive cumsum of x2 at i
      float E = fminf(inv4b2 * acc2, 100.0f);
      float err = expf(E - R) - expf(-R);
      if (STORE) {
        float ec = fminf(err, 100.0f);
        out[j] = ec;
        ssum += ec;
      } else {
        m = fmaxf(m, err);
      }
    }
  }
  if (STORE) return wave_sum_wmma(ssum);  // EXEC is full here (no masking)
  return wave_max_f(m);
}

// ---------------- Tensor Data Mover: 1D LDS -> global store ----------------
// amdgpu-toolchain (clang-23) 6-arg builtin form:
//   (uint32x4 g0, int32x8 g1, int32x4 g2, int32x4 g3, int32x8 pad, i32 cpol)
// D# per CDNA5 ISA:
// group0: count=1, lds_addr, global_addr[56:0], type=2.
// group1: data_size=4B, tensor_dim0=nelem, tensor_dim1=1, tile_dim0=nelem,
//         tile_dim1=1, strides=nelem.
__device__ __forceinline__ void tdm_store_1d(unsigned lds_byte_addr,
                                             const float* gptr,
                                             unsigned nelem) {
  uint64_t ga = (uint64_t)(uintptr_t)gptr;
  v4u g0;
  g0[0] = 1u;                                            // count=1, user D#
  g0[1] = lds_byte_addr;                                 // lds_addr
  g0[2] = (unsigned)(ga & 0xffffffffu);                  // global_addr lo
  g0[3] = (unsigned)((ga >> 32) & 0x01ffffffu) | (2u << 30);  // hi | type=2
  v8i g1;
  g1[0] = (int)(2u << 16);                               // data_size = 4B
  g1[1] = (int)((nelem & 0xffffu) << 16);                // tensor_dim0 lo16
  g1[2] = (int)((nelem >> 16) | (1u << 16));             // td0 hi16 | td1 lo16=1
  g1[3] = (int)((nelem & 0xffffu) << 16);                // td1 hi16=0 | tile_dim0
  g1[4] = 1;                                             // tile_dim1=1, tile_dim2=0
  g1[5] = (int)nelem;                                    // td0_stride lo32
  g1[6] = (int)((nelem & 0xffffu) << 16);                // td0_stride hi=0 | td1_stride lo16
  g1[7] = (int)(nelem >> 16);                            // td1_stride hi
  v4i z4 = {0, 0, 0, 0};
  v8i z8 = {0, 0, 0, 0, 0, 0, 0, 0};
  __builtin_amdgcn_tensor_store_from_lds(g0, g1, z4, z4, z8, 0);
}

// ---------------- main kernel ----------------------------------------------

__global__ __launch_bounds__(kThreads) void volsdf_sampler_kernel(
    const float* __restrict__ cam_loc, const float* __restrict__ ray_dirs,
    const float* __restrict__ smin_in, const float* __restrict__ smax_in,
    const float* __restrict__ beta_in, float* __restrict__ out) {
  // Double-buffered per-ray arrays. 5 up-samples => final data in the B set.
  __shared__ float intA[kRaysPerBlock][kMaxNs];
  __shared__ float sdfA[kRaysPerBlock][kMaxNs];
  __shared__ float intB[kRaysPerBlock][kMaxNs];
  __shared__ float sdfB[kRaysPerBlock][kMaxNs];

  const int wid  = threadIdx.x >> 5;
  const int lane = threadIdx.x & 31;
  const int ray  = blockIdx.x * kRaysPerBlock + wid;

  const float cx = cam_loc[0], cy = cam_loc[1], cz = cam_loc[2];
  const float rx = ray_dirs[ray * 3 + 0];
  const float ry = ray_dirs[ray * 3 + 1];
  const float rz = ray_dirs[ray * 3 + 2];
  const float smin = smin_in[ray];
  const float smax = smax_in[ray];
  const float beta_t = beta_in[0];

  float* intCur = &intA[wid][0];
  float* sdfCur = &sdfA[wid][0];
  float* intNxt = &intB[wid][0];
  float* sdfNxt = &sdfB[wid][0];

  // beta_p = smax / 2 / sqrt((N-1) * log(1+eps))
  float beta_p = (smax * 0.5f) / sqrtf((float)(kN - 1) * logf(1.0f + kEps));

  // ---- iteration 0: uniform sampling + sphere SDF ----
  for (int j = lane; j < kN; j += 32) {
    float t = smin + ((float)j / (float)(kN - 1)) * (smax - smin);
    intCur[j] = t;
    float px = cx + t * rx, py = cy + t * ry, pz = cz + t * rz;
    sdfCur[j] = sqrtf(px * px + py * py + pz * pz) - 1.0f;
  }
  __syncthreads();

  int Ns = kN;
  bool not_valid = false;
  float dlt[kMaxCh], s0c[kMaxCh], dsc[kMaxCh];
  int NI = 0, CH = 0, base = 0;

  for (int itn = 0; itn <= kAlgoIters; ++itn) {
    if (itn > 0) {
      // ================= upsample (uses previous iteration's caches) ======
      float frac[kMaxCh];
      int nbv[kMaxCh], cnts[kMaxCh];
      float total =
          eval_error<true>(beta_p, NI, CH, base, lane, dlt, s0c, dsc, frac);
      int snb = 0;
#pragma unroll
      for (int j = 0; j < kMaxCh; ++j) {
        int i = base + j;
        if ((j < CH) && (i < NI)) {
          float ep = (float)kN * frac[j] / (total + 1e-6f);
          float fl = floorf(ep);
          nbv[j] = (int)fl;
          frac[j] = ep - fl;
          snb += nbv[j];
        } else {
          nbv[j] = 0;
          frac[j] = -1.0f;
        }
      }
      int sum_nb = wave_sum_i(snb);
      int kk = (NI < kN) ? NI : kN;           // k = min(Ns-1, N)
      int remaining = kN - sum_nb;
      int rounds = (remaining < kk) ? remaining : kk;
      if (rounds < 0) rounds = 0;
      // top-k fractional insert; ties broken toward the lowest index,
      // matching lax.top_k ordering.
      for (int r = 0; r < rounds; ++r) {
        float bv = -3.4e38f;
        int bi = 0x7fffffff;
#pragma unroll
        for (int j = 0; j < kMaxCh; ++j) {
          int i = base + j;
          if ((j < CH) && (i < NI) && (frac[j] > bv)) { bv = frac[j]; bi = i; }
        }
#pragma unroll
        for (int off = 16; off >= 1; off >>= 1) {
          float ov = __shfl_xor(bv, off, 32);
          int oi = __shfl_xor(bi, off, 32);
          if (ov > bv || (ov == bv && oi < bi)) { bv = ov; bi = oi; }
        }
        if (bi >= base && bi < base + CH) {   // owner lane updates
          int j = bi - base;
          nbv[j] += 1;
          frac[j] = -2.0f;
        }
      }
      if (lane == 0) nbv[0] += kN - (sum_nb + rounds);  // leftover fixup
      // counts + exclusive scan -> output start positions
      int lt = 0;
#pragma unroll
      for (int j = 0; j < kMaxCh; ++j) {
        int i = base + j;
        int c = ((j < CH) && (i < NI)) ? (nbv[j] + 1) : 0;
        cnts[j] = c;
        lt += c;
      }
      int pos = wave_excl_scan_i(lt, lane);
      // repeat-interleave expansion
#pragma unroll
      for (int j = 0; j < kMaxCh; ++j) {
        int i = base + j;
        if ((j < CH) && (i < NI)) {
          float t0 = intCur[i];
          float s0 = s0c[j];
          float s1 = sdfCur[i + 1];
          float d = dlt[j];
          float denom = (float)nbv[j] + 1.0f;
          for (int o = 0; o < cnts[j]; ++o) {
            float of = (float)o;
            float ti = t0 + of * d / denom;
            float si = s0 + of * (s1 - s0) / denom;
            if (o > 0 && not_valid) {  // recompute SDF at inserted samples
              float px = cx + ti * rx, py = cy + ti * ry, pz = cz + ti * rz;
              si = sqrtf(px * px + py * py + pz * pz) - 1.0f;
            }
            intNxt[pos + o] = ti;
            sdfNxt[pos + o] = si;
          }
          pos += cnts[j];
        }
      }
      if (lane == 0) {  // carry over the final endpoint
        intNxt[NI + kN] = intCur[NI];
        sdfNxt[NI + kN] = sdfCur[NI];
      }
      __syncthreads();
      float* tp;
      tp = intCur; intCur = intNxt; intNxt = tp;
      tp = sdfCur; sdfCur = sdfNxt; sdfNxt = tp;
      Ns += kN;
    }

    // ================= rebuild per-lane caches for current Ns =============
    NI = Ns - 1;
    CH = (NI + 31) >> 5;
    base = lane * CH;
#pragma unroll
    for (int j = 0; j < kMaxCh; ++j) {
      int i = base + j;
      if ((j < CH) && (i < NI)) {
        float s0 = sdfCur[i], s1 = sdfCur[i + 1];
        float d = intCur[i + 1] - intCur[i];
        dlt[j] = d;
        s0c[j] = s0;
        float ds = fmaxf(0.0f, (fabsf(s1) + fabsf(s0) - d) * 0.5f);
        float g0 = (s0 > 0.0f) ? 1.0f : ((s0 < 0.0f) ? -1.0f : 0.0f);
        float g1 = (s1 > 0.0f) ? 1.0f : ((s1 < 0.0f) ? -1.0f : 0.0f);
        dsc[j] = (g0 != g1) ? 0.0f : ds;
      } else {
        dlt[j] = 0.0f; s0c[j] = 0.0f; dsc[j] = 0.0f;
      }
    }

    // ================= find_beta_p ========================================
    float errT =
        eval_error<false>(beta_t, NI, CH, base, lane, dlt, s0c, dsc, nullptr);
    float errP =
        eval_error<false>(beta_p, NI, CH, base, lane, dlt, s0c, dsc, nullptr);
    bool already_ok = (errT <= kEps);
    bool p_not_ok = (errP > kEps);
    bool not_fin = (!already_ok) && (!p_not_ok);
    float lo = beta_t, hi = beta_p;
    for (int b = 0; b < kBisect; ++b) {  // full-batch bisection, as reference
      float mid = 0.5f * (hi + lo);
      float em =
          eval_error<false>(mid, NI, CH, base, lane, dlt, s0c, dsc, nullptr);
      if (em > kEps) lo = mid; else hi = mid;
    }
    float res = 1.0f;
    if (not_fin) res = hi;
    if (already_ok) res = beta_t;
    if (p_not_ok) res = beta_p;
    beta_p = res;
    not_valid = !already_ok;
  }

  // ================= outputs ==============================================
  if (lane == 0) {
    out[(size_t)2 * kNp * kMaxNs + ray] = beta_p;
  }
  __syncthreads();  // all rays' final tiles resident in the B buffers

  if (threadIdx.x < 32) {  // one wave issues the TDM tile stores
    const size_t rayBase = (size_t)blockIdx.x * kRaysPerBlock;
    unsigned ldsInt = (unsigned)(uintptr_t)&intB[0][0];
    unsigned ldsSdf = (unsigned)(uintptr_t)&sdfB[0][0];
    const unsigned nelem = (unsigned)(kRaysPerBlock * kMaxNs);  // 3072 f32
    tdm_store_1d(ldsInt, out + rayBase * kMaxNs, nelem);
    tdm_store_1d(ldsSdf, out + (size_t)kNp * kMaxNs + rayBase * kMaxNs, nelem);
    __builtin_amdgcn_s_wait_tensorcnt(0);
  }
}

// ---------------------------------------------------------------------------

extern "C" void kernel_launch(void* const* d_in, const int* in_sizes, int n_in,
                              void* d_out, int out_size, void* d_ws,
                              size_t ws_size, hipStream_t stream) {
  (void)in_sizes; (void)n_in; (void)out_size; (void)d_ws; (void)ws_size;
  const float* cam  = (const float*)d_in[0];
  const float* rdir = (const float*)d_in[1];
  const float* smin = (const float*)d_in[2];
  const float* smax = (const float*)d_in[3];
  const float* beta = (const float*)d_in[4];
  float* out = (float*)d_out;

  dim3 grid(kNp / kRaysPerBlock);
  dim3 block(kThreads);
  volsdf_sampler_kernel<<<grid, block, 0, stream>>>(cam, rdir, smin, smax,
                                                    beta, out);
}